// HybridQCNNGraphAttention_65481071396973
// MI455X (gfx1250) — compile-verified
//
#include <hip/hip_runtime.h>
#include <hip/hip_bf16.h>
#include <math.h>

typedef __attribute__((ext_vector_type(2))) float v2f;
typedef __attribute__((ext_vector_type(8))) float v8f;

#define N_NODES 8192
#define TILE 16
#define NT (N_NODES / TILE)      // 512 tiles
#define WPB 8                    // waves per block (one 16-row i-tile each)
#define THREADS 256

// Per-wave LDS regions + in-order LDS within a wave => only a compiler
// reordering fence is needed (zero instructions), no block barrier.
__device__ __forceinline__ void wave_lds_fence() { asm volatile("" ::: "memory"); }

// ---------------------------------------------------------------- normalize
__global__ __launch_bounds__(THREADS)
void k_normalize(const float* __restrict__ x, float* __restrict__ xn) {
    int i = blockIdx.x * blockDim.x + threadIdx.x;
    if (i >= N_NODES) return;
    float v[8]; float s = 0.f;
#pragma unroll
    for (int d = 0; d < 8; ++d) { v[d] = x[i * 8 + d]; s += v[d] * v[d]; }
    float inv = 1.0f / (sqrtf(s) + 1e-12f);
#pragma unroll
    for (int d = 0; d < 8; ++d) xn[i * 8 + d] = v[d] * inv;
}

// --------------------- fused adjacency + diffusion + GNN + q/k projection
// neighbour_sum[i] = sum_j [ (xn_i.xn_j)^2 >= 0.8 ] x_j  -  x_i   (diag always passes)
// diffused = relu(neighbour_sum @ gnn_w.T + gnn_b)
// q = diffused @ rot * (log2(e)/sqrt(8))   [log-base-2 softmax domain]
// k = diffused @ ent
__global__ __launch_bounds__(THREADS)
void k_diffuse(const float* __restrict__ xn, const float* __restrict__ x,
               const float* __restrict__ gnn_w, const float* __restrict__ gnn_b,
               const float* __restrict__ rot, const float* __restrict__ ent,
               float* __restrict__ diffused, float* __restrict__ qout,
               float* __restrict__ kout) {
    __shared__ float ldsM[WPB][16][16];
    __shared__ float ldsS[WPB][16][8];
    __shared__ float ldsD[WPB][16][8];
    const int wave = threadIdx.x >> 5;
    const int lane = threadIdx.x & 31;
    const int hf   = lane >> 4;
    const int lm   = lane & 15;
    const int l8   = lane & 7;
    const bool low = lm < 8;
    const int i0   = (blockIdx.x * WPB + wave) * TILE;

    // A-layout K-chunks of this wave's 16 normalized rows
    v2f aq[2];
#pragma unroll
    for (int c = 0; c < 2; ++c) {
        const float* p = xn + (i0 + lm) * 8 + 4 * c + 2 * hf;
        aq[c][0] = p[0]; aq[c][1] = p[1];
    }

    v8f acc = {};
    for (int j0 = 0; j0 < N_NODES; j0 += TILE) {
        v2f b0, b1;     // B-layout chunks of xn_j^T (same gather as A, rows j0+lm)
        { const float* p = xn + (j0 + lm) * 8 + 2 * hf;     b0[0] = p[0]; b0[1] = p[1]; }
        { const float* p = xn + (j0 + lm) * 8 + 4 + 2 * hf; b1[0] = p[0]; b1[1] = p[1]; }
        v8f g = {};
        g = __builtin_amdgcn_wmma_f32_16x16x4_f32(false, aq[0], false, b0, (short)0, g, false, false);
        g = __builtin_amdgcn_wmma_f32_16x16x4_f32(false, aq[1], false, b1, (short)0, g, false, false);
#pragma unroll
        for (int r = 0; r < 8; ++r)
            ldsM[wave][r + 8 * hf][lm] = (g[r] * g[r] >= 0.8f) ? 1.0f : 0.0f;
        wave_lds_fence();
        // acc += M(16x16) @ X_j(16x8) as 4 chained K=4 WMMAs
#pragma unroll
        for (int c = 0; c < 4; ++c) {
            v2f ma;
            ma[0] = ldsM[wave][lm][4 * c + 2 * hf + 0];
            ma[1] = ldsM[wave][lm][4 * c + 2 * hf + 1];
            int k0 = j0 + 4 * c + 2 * hf;
            float v0 = x[(k0 + 0) * 8 + l8];     // in-bounds for all lanes
            float v1 = x[(k0 + 1) * 8 + l8];
            v2f bx;
            bx[0] = low ? v0 : 0.0f;
            bx[1] = low ? v1 : 0.0f;
            acc = __builtin_amdgcn_wmma_f32_16x16x4_f32(false, ma, false, bx, (short)0, acc, false, false);
        }
        wave_lds_fence();
    }

    // subtract self-contribution, stage rows
    if (low) {
#pragma unroll
        for (int r = 0; r < 8; ++r)
            ldsS[wave][r + 8 * hf][lm] = acc[r] - x[(i0 + r + 8 * hf) * 8 + lm];
    }
    wave_lds_fence();
    // GNN 8x8 linear + relu, 4 outputs per lane
#pragma unroll
    for (int p = 0; p < 4; ++p) {
        int t = lane * 4 + p, il = t >> 3, e = t & 7;
        float s = gnn_b[e];
#pragma unroll
        for (int d = 0; d < 8; ++d) s = fmaf(ldsS[wave][il][d], gnn_w[e * 8 + d], s);
        float dv = fmaxf(s, 0.0f);
        diffused[(i0 + il) * 8 + e] = dv;
        ldsD[wave][il][e] = dv;
    }
    wave_lds_fence();
    // q/k projection; q pre-scaled by log2(e)/sqrt(8) => softmax via exp2
    const float scale = 1.4426950408889634f * 0.3535533905932738f;
#pragma unroll
    for (int p = 0; p < 4; ++p) {
        int t = lane * 4 + p, il = t >> 3, e = t & 7;
        float sq = 0.f, sk = 0.f;
#pragma unroll
        for (int d = 0; d < 8; ++d) {
            float h = ldsD[wave][il][d];
            sq = fmaf(h, rot[d * 8 + e], sq);
            sk = fmaf(h, ent[d * 8 + e], sk);
        }
        qout[(i0 + il) * 8 + e] = sq * scale;
        kout[(i0 + il) * 8 + e] = sk;
    }
}

// --------------------- two-pass flash attention (row sum via ones-column)
__global__ __launch_bounds__(THREADS)
void k_attn(const float* __restrict__ q, const float* __restrict__ kv,
            const float* __restrict__ diff, float* __restrict__ attended) {
    __shared__ float ldsP[WPB][16][16];
    const int wave = threadIdx.x >> 5;
    const int lane = threadIdx.x & 31;
    const int hf   = lane >> 4;
    const int lm   = lane & 15;
    const int l8   = lane & 7;
    const bool low = lm < 8;
    const float tail = (lm == 8) ? 1.0f : 0.0f;   // ones-column constant, hoisted
    const int i0   = (blockIdx.x * WPB + wave) * TILE;

    v2f aq[2];
#pragma unroll
    for (int c = 0; c < 2; ++c) {
        const float* p = q + (i0 + lm) * 8 + 4 * c + 2 * hf;
        aq[c][0] = p[0]; aq[c][1] = p[1];
    }

    // ---- pass 1: global row max (2 WMMA + 8 fmax per tile)
    v8f mt;
#pragma unroll
    for (int r = 0; r < 8; ++r) mt[r] = -__builtin_inff();
    for (int j0 = 0; j0 < N_NODES; j0 += TILE) {
        v2f b0, b1;
        { const float* p = kv + (j0 + lm) * 8 + 2 * hf;     b0[0] = p[0]; b0[1] = p[1]; }
        { const float* p = kv + (j0 + lm) * 8 + 4 + 2 * hf; b1[0] = p[0]; b1[1] = p[1]; }
        v8f L = {};
        L = __builtin_amdgcn_wmma_f32_16x16x4_f32(false, aq[0], false, b0, (short)0, L, false, false);
        L = __builtin_amdgcn_wmma_f32_16x16x4_f32(false, aq[1], false, b1, (short)0, L, false, false);
#pragma unroll
        for (int r = 0; r < 8; ++r) mt[r] = fmaxf(mt[r], L[r]);
    }
    float mx[8];
#pragma unroll
    for (int r = 0; r < 8; ++r) {          // one butterfly total, not per tile
        float t = mt[r];
#pragma unroll
        for (int off = 1; off < 16; off <<= 1) t = fmaxf(t, __shfl_xor(t, off, 32));
        mx[r] = t;
    }

    // ---- pass 2: P = exp2(L - mx); acc += P @ [diffused_j | 1] (sum in col 8)
    v8f acc = {};
    for (int j0 = 0; j0 < N_NODES; j0 += TILE) {
        v2f b0, b1;
        { const float* p = kv + (j0 + lm) * 8 + 2 * hf;     b0[0] = p[0]; b0[1] = p[1]; }
        { const float* p = kv + (j0 + lm) * 8 + 4 + 2 * hf; b1[0] = p[0]; b1[1] = p[1]; }
        v8f L = {};
        L = __builtin_amdgcn_wmma_f32_16x16x4_f32(false, aq[0], false, b0, (short)0, L, false, false);
        L = __builtin_amdgcn_wmma_f32_16x16x4_f32(false, aq[1], false, b1, (short)0, L, false, false);
#pragma unroll
        for (int r = 0; r < 8; ++r)
            ldsP[wave][r + 8 * hf][lm] = exp2f(L[r] - mx[r]);   // raw v_exp_f32
        wave_lds_fence();
#pragma unroll
        for (int c = 0; c < 4; ++c) {
            v2f pa;
            pa[0] = ldsP[wave][lm][4 * c + 2 * hf + 0];
            pa[1] = ldsP[wave][lm][4 * c + 2 * hf + 1];
            int k0 = j0 + 4 * c + 2 * hf;
            float v0 = diff[(k0 + 0) * 8 + l8];
            float v1 = diff[(k0 + 1) * 8 + l8];
            v2f bd;   // cols 0..7 = diffused, col 8 = 1.0 (row-sum accumulator)
            bd[0] = low ? v0 : tail;
            bd[1] = low ? v1 : tail;
            acc = __builtin_amdgcn_wmma_f32_16x16x4_f32(false, pa, false, bd, (short)0, acc, false, false);
        }
        wave_lds_fence();
    }
    // attended = acc[:,0:8] / acc[:,8]
#pragma unroll
    for (int r = 0; r < 8; ++r) {
        float s = __shfl(acc[r], 8 + 16 * hf, 32);
        if (low) attended[(i0 + r + 8 * hf) * 8 + lm] = acc[r] / s;
    }
}

// ------------------------------------------------------------- QCNN MLP head
__global__ __launch_bounds__(THREADS)
void k_qcnn(const float* __restrict__ att,
            const float* __restrict__ w0, const float* __restrict__ c0,
            const float* __restrict__ w1, const float* __restrict__ c1,
            const float* __restrict__ w2, const float* __restrict__ c2,
            const float* __restrict__ w3, const float* __restrict__ c3,
            const float* __restrict__ w4, const float* __restrict__ c4,
            const float* __restrict__ w5, const float* __restrict__ c5,
            float* __restrict__ out) {
    int i = blockIdx.x * blockDim.x + threadIdx.x;
    if (i >= N_NODES) return;
    float a[16], b[16];
#pragma unroll
    for (int d = 0; d < 8; ++d) a[d] = att[i * 8 + d];
#pragma unroll
    for (int o = 0; o < 16; ++o) {           // 8 -> 16
        float s = c0[o];
#pragma unroll
        for (int d = 0; d < 8; ++d) s = fmaf(w0[o * 8 + d], a[d], s);
        b[o] = tanhf(s);
    }
#pragma unroll
    for (int o = 0; o < 16; ++o) {           // 16 -> 16
        float s = c1[o];
#pragma unroll
        for (int d = 0; d < 16; ++d) s = fmaf(w1[o * 16 + d], b[d], s);
        a[o] = tanhf(s);
    }
#pragma unroll
    for (int o = 0; o < 12; ++o) {           // 16 -> 12
        float s = c2[o];
#pragma unroll
        for (int d = 0; d < 16; ++d) s = fmaf(w2[o * 16 + d], a[d], s);
        b[o] = tanhf(s);
    }
#pragma unroll
    for (int o = 0; o < 8; ++o) {            // 12 -> 8
        float s = c3[o];
#pragma unroll
        for (int d = 0; d < 12; ++d) s = fmaf(w3[o * 12 + d], b[d], s);
        a[o] = tanhf(s);
    }
#pragma unroll
    for (int o = 0; o < 4; ++o) {            // 8 -> 4
        float s = c4[o];
#pragma unroll
        for (int d = 0; d < 8; ++d) s = fmaf(w4[o * 8 + d], a[d], s);
        b[o] = tanhf(s);
    }
    float s = c5[0];                         // 4 -> 1, sigmoid
#pragma unroll
    for (int d = 0; d < 4; ++d) s = fmaf(w5[d], b[d], s);
    out[i] = 1.0f / (1.0f + __expf(-s));
}

// ---------------------------------------------------------------------------
extern "C" void kernel_launch(void* const* d_in, const int* in_sizes, int n_in,
                              void* d_out, int out_size, void* d_ws, size_t ws_size,
                              hipStream_t stream) {
    const float* inputs   = (const float*)d_in[0];
    const float* rotation = (const float*)d_in[1];
    const float* entangle = (const float*)d_in[2];
    const float* gnn_w    = (const float*)d_in[3];
    const float* gnn_b    = (const float*)d_in[4];
    const float* qw[6]; const float* qb[6];
    for (int l = 0; l < 6; ++l) {
        qw[l] = (const float*)d_in[5 + 2 * l];
        qb[l] = (const float*)d_in[6 + 2 * l];
    }
    float* ws       = (float*)d_ws;
    float* xn       = ws;                  // 64K floats; reused as `attended`
    float* diffused = ws + 1 * 65536;
    float* q        = ws + 2 * 65536;
    float* k        = ws + 3 * 65536;
    float* attended = xn;                  // xn dead after k_diffuse

    k_normalize<<<N_NODES / THREADS, THREADS, 0, stream>>>(inputs, xn);
    k_diffuse<<<NT / WPB, THREADS, 0, stream>>>(xn, inputs, gnn_w, gnn_b,
                                                rotation, entangle, diffused, q, k);
    k_attn<<<NT / WPB, THREADS, 0, stream>>>(q, k, diffused, attended);
    k_qcnn<<<N_NODES / THREADS, THREADS, 0, stream>>>(attended,
        qw[0], qb[0], qw[1], qb[1], qw[2], qb[2],
        qw[3], qb[3], qw[4], qb[4], qw[5], qb[5],
        (float*)d_out);
}